// QuaternionPositionToMatrixLayer_72378788873083
// MI455X (gfx1250) — compile-verified
//
#include <hip/hip_runtime.h>

// Quaternion+translation -> 3x4 pose matrix.
// Memory-bound: 76 B/row, ~0.4 FLOP/B -> pure bandwidth problem on MI455X
// (304 MB @ 23.3 TB/s ~= 13 us). WMMA cannot help; the win is the CDNA5
// data-movement path: async global->LDS B128 staging of the 28B/row AoS input
// (ASYNCcnt), conflict-free stride-7 LDS reads, and 3x B128 non-temporal
// global stores for the 48B/row (16B-aligned) output.

typedef float v4f __attribute__((ext_vector_type(4)));
typedef int   v4i __attribute__((ext_vector_type(4)));

#define BLOCK      256
#define ROWS_PB    256                 // rows per block (1 row / thread)
#define IN_FLOATS  (ROWS_PB * 7)       // 1792 floats = 7168 B (16B-aligned per block)

#if defined(__gfx1250__) && __has_builtin(__builtin_amdgcn_global_load_async_to_lds_b128)
  #define USE_ASYNC 1
#else
  #define USE_ASYNC 0
#endif

#if USE_ASYNC
typedef __attribute__((address_space(1))) v4i gv4i;   // global int4
typedef __attribute__((address_space(3))) v4i lv4i;   // LDS int4
#endif

__global__ __launch_bounds__(BLOCK)
void quat_pose_kernel(const float* __restrict__ x, float* __restrict__ out, int n) {
  __shared__ float s_in[IN_FLOATS];

  const int tid = threadIdx.x;
  const long long base = (long long)blockIdx.x * IN_FLOATS;   // float index of tile start
  const long long n7   = (long long)n * 7LL;
  const long long rem  = n7 - base;
  const int nf  = (rem < (long long)IN_FLOATS) ? (int)rem : IN_FLOATS; // floats staged
  const int nv4 = nf >> 2;                                             // full float4 chunks

#if USE_ASYNC
  // CDNA5 async DMA: global -> LDS, 16B per lane, no VGPR round trip (ASYNCcnt).
  for (int i = tid; i < nv4; i += BLOCK) {
    gv4i* g = (gv4i*)(unsigned long long)(x + base + (long long)i * 4);
    lv4i* l = (lv4i*)(unsigned)(unsigned long long)(&s_in[i * 4]);
    __builtin_amdgcn_global_load_async_to_lds_b128(g, l, /*offset=*/0, /*cpol=*/0);
  }
#else
  // Fallback: vectorized load->LDS through VGPRs.
  for (int i = tid; i < nv4; i += BLOCK) {
    v4f v = *(const v4f*)(x + base + (long long)i * 4);
    *(v4f*)(&s_in[i * 4]) = v;
  }
#endif
  // Scalar tail (only for the final partial block).
  for (int i = (nv4 << 2) + tid; i < nf; i += BLOCK)
    s_in[i] = x[base + i];

#if USE_ASYNC
  #if __has_builtin(__builtin_amdgcn_s_wait_asynccnt)
  __builtin_amdgcn_s_wait_asynccnt(0);
  #else
  asm volatile("s_wait_asynccnt 0" ::: "memory");
  #endif
#endif
  __syncthreads();

  const long long row = (long long)blockIdx.x * ROWS_PB + tid;
  if (row < n) {
    // Stride-7 LDS reads: odd stride -> conflict-free across 64 banks.
    const float* q = &s_in[tid * 7];
    const float q0 = q[0], q1 = q[1], q2 = q[2], q3 = q[3];
    const float t0 = q[4], t1 = q[5], t2 = q[6];

    const float q00 = q0 * q0, q11 = q1 * q1, q22 = q2 * q2, q33 = q3 * q3;
    const float q01 = q0 * q1, q02 = q0 * q2, q03 = q0 * q3;
    const float q12 = q1 * q2, q13 = q1 * q3, q23 = q2 * q3;

    v4f r0 = { q00 + q11 - q22 - q33, 2.0f * (q12 - q03), 2.0f * (q13 + q02), t0 };
    v4f r1 = { 2.0f * (q12 + q03), q00 - q11 + q22 - q33, 2.0f * (q23 - q01), t1 };
    v4f r2 = { 2.0f * (q13 - q02), 2.0f * (q23 + q01), q00 - q11 - q22 + q33, t2 };

    // 48 B/row, 16B-aligned -> three B128 stores; NT hint: output never reused.
    v4f* o = (v4f*)(out + row * 12);
    __builtin_nontemporal_store(r0, o + 0);
    __builtin_nontemporal_store(r1, o + 1);
    __builtin_nontemporal_store(r2, o + 2);
  }
}

extern "C" void kernel_launch(void* const* d_in, const int* in_sizes, int n_in,
                              void* d_out, int out_size, void* d_ws, size_t ws_size,
                              hipStream_t stream) {
  (void)n_in; (void)out_size; (void)d_ws; (void)ws_size;
  const float* x = (const float*)d_in[0];
  float* out = (float*)d_out;
  const int n = in_sizes[0] / 7;                      // number of rows (B)
  const int blocks = (n + ROWS_PB - 1) / ROWS_PB;     // 15625 for B = 4e6
  quat_pose_kernel<<<blocks, BLOCK, 0, stream>>>(x, out, n);
}